// GCN_2826088481300
// MI455X (gfx1250) — compile-verified
//
#include <hip/hip_runtime.h>
#include <math.h>

typedef float v2f __attribute__((ext_vector_type(2)));
typedef float v8f __attribute__((ext_vector_type(8)));

#define NGRAPH 512

// ---------------------------------------------------------------------------
// Degree / normalization
// ---------------------------------------------------------------------------
__global__ void deg_init(float* __restrict__ deg, int n) {
    int i = blockIdx.x * blockDim.x + threadIdx.x;
    if (i < n) deg[i] = 1.0f;  // self-loop contribution
}

__global__ void deg_edge(const int* __restrict__ dst, float* __restrict__ deg, int e) {
    int i = blockIdx.x * blockDim.x + threadIdx.x;
    if (i < e) atomicAdd(&deg[dst[i]], 1.0f);
}

__global__ void deg_to_dinv(float* __restrict__ deg, int n) {
    int i = blockIdx.x * blockDim.x + threadIdx.x;
    if (i < n) deg[i] = rsqrtf(deg[i]);  // deg >= 1 always (self loops)
}

// ---------------------------------------------------------------------------
// fp32 WMMA GEMM: out[N,F] = (RELU_IN ? relu(A) : A)[N,K] @ W[K,F]
// One wave computes a 16x32 output tile: two V_WMMA_F32_16X16X4_F32
// accumulators sharing one A fragment. A fetched as float4 (b128) covering
// K..K+3 of the lane's row; lane-half selects its (K-pair) per ISA layout.
// N multiple of 16, F multiple of 32, K multiple of 4 (true for all layers).
// ---------------------------------------------------------------------------
template <bool RELU_IN>
__global__ __launch_bounds__(256) void gcn_gemm_wmma_f32(
    const float* __restrict__ A, const float* __restrict__ W,
    float* __restrict__ out, int N, int K, int F) {
    const int lane = threadIdx.x & 31;
    const int wave = threadIdx.x >> 5;
    const int colTiles = F >> 5;  // 32-wide column tiles
    const long nTiles = (long)(N >> 4) * colTiles;
    const long tile = (long)blockIdx.x * (blockDim.x >> 5) + wave;
    if (tile >= nTiles) return;  // wave-uniform: EXEC stays all-ones for WMMA

    const int row0 = (int)(tile / colTiles) << 4;
    const int col0 = (int)(tile % colTiles) << 5;
    const int m = lane & 15;
    const int hf = lane >> 4;      // 0: K pair {0,1}; 1: K pair {2,3}
    const int khalf = hf << 1;

    const float4* arow = (const float4*)(A + (size_t)(row0 + m) * K);
    const float* wc = W + (size_t)khalf * F + col0 + m;

    v8f c0 = {};
    v8f c1 = {};
    const int ksteps = K >> 2;
    for (int k4 = 0; k4 < ksteps; ++k4) {
        float4 aq = arow[k4];
        float a0 = hf ? aq.z : aq.x;
        float a1 = hf ? aq.w : aq.y;
        if (RELU_IN) { a0 = fmaxf(a0, 0.0f); a1 = fmaxf(a1, 0.0f); }
        v2f av; av.x = a0; av.y = a1;
        const float* w0 = wc + (size_t)(k4 << 2) * F;
        v2f b0; b0.x = w0[0];  b0.y = w0[F];
        v2f b1; b1.x = w0[16]; b1.y = w0[F + 16];
        // (neg_a, A, neg_b, B, c_mod, C, reuse_a, reuse_b)
        c0 = __builtin_amdgcn_wmma_f32_16x16x4_f32(false, av, false, b0,
                                                   (short)0, c0, false, false);
        c1 = __builtin_amdgcn_wmma_f32_16x16x4_f32(false, av, false, b1,
                                                   (short)0, c1, false, false);
    }

    // D layout: VGPR v -> row M = v + 8*(lane>>4), col N = lane&15
    const int mo = hf << 3;
    float* orow = out + (size_t)(row0 + mo) * F + col0 + m;
#pragma unroll
    for (int v = 0; v < 8; ++v) {
        orow[(size_t)v * F] = c0[v];
        orow[(size_t)v * F + 16] = c1[v];
    }
}

// ---------------------------------------------------------------------------
// out[i,f] = bias[f] + xw[i,f] * dinv[i]^2   (bias + self-loop edge folded in)
// ---------------------------------------------------------------------------
__global__ void init_agg(const float* __restrict__ xw, const float* __restrict__ dinv,
                         const float* __restrict__ bias, float* __restrict__ out,
                         int N, int F) {
    long idx = (long)blockIdx.x * blockDim.x + threadIdx.x;
    if (idx >= (long)N * F) return;
    int i = (int)(idx / F);
    int f = (int)(idx % F);
    float di = dinv[i];
    out[idx] = bias[f] + xw[idx] * di * di;
}

// ---------------------------------------------------------------------------
// For each edge e: out[dst[e], :] += xw[src[e], :] * dinv[src]*dinv[dst]
// Grid-stride; one item = (edge, 4 features). Next iteration's gather row is
// prefetched (global_prefetch) to overlap the index->row dependency chain.
// cshift = log2(F/4).
// ---------------------------------------------------------------------------
__global__ void edge_scatter(const int* __restrict__ src, const int* __restrict__ dst,
                             const float* __restrict__ dinv, const float* __restrict__ xw,
                             float* __restrict__ out, int E, int F, int cshift) {
    const long total = (long)E << cshift;
    const long stride = (long)gridDim.x * blockDim.x;
    const int cmask = (1 << cshift) - 1;
    for (long idx = (long)blockIdx.x * blockDim.x + threadIdx.x; idx < total;
         idx += stride) {
        long nidx = idx + stride;
        if (nidx < total) {
            int ne = (int)(nidx >> cshift);
            int nf4 = ((int)(nidx & cmask)) << 2;
            __builtin_prefetch(xw + (size_t)src[ne] * F + nf4, 0, 0);
        }
        int e = (int)(idx >> cshift);
        int f4 = ((int)(idx & cmask)) << 2;
        int s = src[e], d = dst[e];
        float nrm = dinv[s] * dinv[d];
        const float4 v = *(const float4*)(xw + (size_t)s * F + f4);
        float* o = out + (size_t)d * F + f4;
        atomicAdd(o + 0, v.x * nrm);
        atomicAdd(o + 1, v.y * nrm);
        atomicAdd(o + 2, v.z * nrm);
        atomicAdd(o + 3, v.w * nrm);
    }
}

// ---------------------------------------------------------------------------
// Pooling: g[batch[i], :] += h[i, :]   (F fixed at 128)
// ---------------------------------------------------------------------------
__global__ void zero_f32(float* __restrict__ p, long n) {
    long i = (long)blockIdx.x * blockDim.x + threadIdx.x;
    if (i < n) p[i] = 0.0f;
}

__global__ void pool_add(const float* __restrict__ h, const int* __restrict__ batch,
                         float* __restrict__ g, int N) {
    long idx = (long)blockIdx.x * blockDim.x + threadIdx.x;
    if (idx >= (long)N * 128) return;
    int i = (int)(idx >> 7);
    int f = (int)(idx & 127);
    atomicAdd(&g[(size_t)batch[i] * 128 + f], h[idx]);
}

// ---------------------------------------------------------------------------
// Head: out = log_softmax(relu(g @ Wl1 + bl1) @ Wl2 + bl2)
// One 64-thread block per graph row.
// ---------------------------------------------------------------------------
__global__ __launch_bounds__(64) void mlp_head(
    const float* __restrict__ g, const float* __restrict__ Wl1,
    const float* __restrict__ bl1, const float* __restrict__ Wl2,
    const float* __restrict__ bl2, float* __restrict__ out) {
    __shared__ float h[64];
    __shared__ float logit[10];
    __shared__ float red[2];
    const int row = blockIdx.x;
    const int t = threadIdx.x;
    const float* gr = g + (size_t)row * 128;

    float acc = bl1[t];
    for (int k = 0; k < 128; ++k) acc += gr[k] * Wl1[k * 64 + t];
    h[t] = fmaxf(acc, 0.0f);
    __syncthreads();

    if (t < 10) {
        float a = bl2[t];
        for (int k = 0; k < 64; ++k) a += h[k] * Wl2[k * 10 + t];
        logit[t] = a;
    }
    __syncthreads();

    if (t == 0) {
        float m = logit[0];
        for (int i = 1; i < 10; ++i) m = fmaxf(m, logit[i]);
        float s = 0.0f;
        for (int i = 0; i < 10; ++i) s += expf(logit[i] - m);
        red[0] = m;
        red[1] = logf(s);
    }
    __syncthreads();

    if (t < 10) out[(size_t)row * 10 + t] = logit[t] - red[0] - red[1];
}

// ---------------------------------------------------------------------------
// Launch
// ---------------------------------------------------------------------------
extern "C" void kernel_launch(void* const* d_in, const int* in_sizes, int n_in,
                              void* d_out, int out_size, void* d_ws, size_t ws_size,
                              hipStream_t stream) {
    (void)n_in; (void)out_size; (void)ws_size;

    const float* x   = (const float*)d_in[0];
    const int*   ei  = (const int*)d_in[1];
    const int*   bat = (const int*)d_in[2];
    const float* W1  = (const float*)d_in[3];
    const float* b1  = (const float*)d_in[4];
    const float* W2  = (const float*)d_in[5];
    const float* b2  = (const float*)d_in[6];
    const float* W3  = (const float*)d_in[7];
    const float* b3  = (const float*)d_in[8];
    const float* Wl1 = (const float*)d_in[9];
    const float* bl1 = (const float*)d_in[10];
    const float* Wl2 = (const float*)d_in[11];
    const float* bl2 = (const float*)d_in[12];

    const int N = in_sizes[2];        // 100000 (multiple of 16)
    const int E = in_sizes[1] / 2;    // 1600000
    const int* src = ei;              // edge_index row 0
    const int* dst = ei + E;          // edge_index row 1

    // Workspace carve-out (256B aligned)
    char* ws = (char*)d_ws;
    size_t off = 0;
    auto carve = [&](size_t bytes) {
        void* p = ws + off;
        off = (off + bytes + 255) & ~(size_t)255;
        return p;
    };
    float* dinv = (float*)carve((size_t)N * 4);          // deg -> dinv in place
    float* bufA = (float*)carve((size_t)N * 128 * 4);    // xw
    float* bufB = (float*)carve((size_t)N * 128 * 4);    // aggregated h
    float* g    = (float*)carve((size_t)NGRAPH * 128 * 4);

    const int B = 256;
    auto cdiv = [](long a, long b) { return (unsigned)((a + b - 1) / b); };
    // grid-stride scatter: ~4 items per thread
    auto sgrid = [&](long total) {
        long blocks = (total + (long)B * 4 - 1) / ((long)B * 4);
        return (unsigned)(blocks < 1 ? 1 : blocks);
    };

    // --- normalization ---
    deg_init<<<cdiv(N, B), B, 0, stream>>>(dinv, N);
    deg_edge<<<cdiv(E, B), B, 0, stream>>>(dst, dinv, E);
    deg_to_dinv<<<cdiv(N, B), B, 0, stream>>>(dinv, N);

    // --- layer 1: 128 -> 32 ---
    gcn_gemm_wmma_f32<false><<<cdiv((long)(N / 16) * (32 / 32), 8), 256, 0, stream>>>(
        x, W1, bufA, N, 128, 32);
    init_agg<<<cdiv((long)N * 32, B), B, 0, stream>>>(bufA, dinv, b1, bufB, N, 32);
    edge_scatter<<<sgrid((long)E * 8), B, 0, stream>>>(src, dst, dinv, bufA, bufB, E, 32, 3);

    // --- layer 2: 32 -> 64 (relu on input) ---
    gcn_gemm_wmma_f32<true><<<cdiv((long)(N / 16) * (64 / 32), 8), 256, 0, stream>>>(
        bufB, W2, bufA, N, 32, 64);
    init_agg<<<cdiv((long)N * 64, B), B, 0, stream>>>(bufA, dinv, b2, bufB, N, 64);
    edge_scatter<<<sgrid((long)E * 16), B, 0, stream>>>(src, dst, dinv, bufA, bufB, E, 64, 4);

    // --- layer 3: 64 -> 128 (relu on input, no relu on output) ---
    gcn_gemm_wmma_f32<true><<<cdiv((long)(N / 16) * (128 / 32), 8), 256, 0, stream>>>(
        bufB, W3, bufA, N, 64, 128);
    init_agg<<<cdiv((long)N * 128, B), B, 0, stream>>>(bufA, dinv, b3, bufB, N, 128);
    edge_scatter<<<sgrid((long)E * 32), B, 0, stream>>>(src, dst, dinv, bufA, bufB, E, 128, 5);

    // --- global add pool ---
    zero_f32<<<cdiv((long)NGRAPH * 128, B), B, 0, stream>>>(g, (long)NGRAPH * 128);
    pool_add<<<cdiv((long)N * 128, B), B, 0, stream>>>(bufB, bat, g, N);

    // --- MLP head + log_softmax ---
    mlp_head<<<NGRAPH, 64, 0, stream>>>(g, Wl1, bl1, Wl2, bl2, (float*)d_out);
}